// AttentionPermMatrix_26465588478183
// MI455X (gfx1250) — compile-verified
//
#include <hip/hip_runtime.h>
#include <math.h>

typedef __attribute__((ext_vector_type(2))) float v2f;
typedef __attribute__((ext_vector_type(8))) float v8f;

#define BLOCKS 64
#define ESIZE  64
#define BSIZE  128              // block_size (pool dim)
#define POS    (ESIZE * BLOCKS) // 4096 contiguous (e, block) positions per slice
#define TEMP_INV (1.0f / 0.7f)
#define EPS    1e-6f
#define SCALE  0.08838834764831845f // 128^-0.5

// ---------------------------------------------------------------------------
// Kernel 1: streaming partial mean-pool over the block_size axis.
// grid (B, 2 [q/k], split), 256 threads. Fully coalesced NT loads:
// thread t accumulates positions {t, t+256, ...} across its s-chunk.
// ---------------------------------------------------------------------------
__global__ __launch_bounds__(256) void perm_pool_partial(
    const float* __restrict__ b_q, const float* __restrict__ b_k,
    float* __restrict__ partial, int split) {
  const int b  = blockIdx.x;
  const int qk = blockIdx.y;
  const int sp = blockIdx.z;
  const int t  = threadIdx.x;

  const float* src    = qk ? b_k : b_q;
  const int    slices = BSIZE / split;
  const float* base   = src + ((size_t)b * BSIZE + (size_t)sp * slices) * POS;

  float acc[16];
#pragma unroll
  for (int i = 0; i < 16; ++i) acc[i] = 0.f;

  for (int s = 0; s < slices; ++s) {
    const float* p = base + (size_t)s * POS;
#pragma unroll
    for (int i = 0; i < 16; ++i)
      acc[i] += __builtin_nontemporal_load(p + t + i * 256);
  }

  float* dst = partial + ((size_t)(b * 2 + qk) * split + sp) * POS;
#pragma unroll
  for (int i = 0; i < 16; ++i) dst[t + i * 256] = acc[i];
}

// ---------------------------------------------------------------------------
// Kernel 2: per-batch fused reduce -> WMMA f32 GEMM -> gumbel/log -> Sinkhorn.
// grid (B), 512 threads = 16 waves; wave w owns C tile (w/4, w%4) of 64x64.
// ---------------------------------------------------------------------------
__global__ __launch_bounds__(512) void perm_sinkhorn(
    const float* __restrict__ b_q, const float* __restrict__ b_k,
    const float* __restrict__ partial, int split,
    const float* __restrict__ gumbel_u, float* __restrict__ out) {
  __shared__ float sq[BLOCKS][ESIZE + 1];   // [block i][e], padded stride 65
  __shared__ float sk[BLOCKS][ESIZE + 1];
  __shared__ float r [BLOCKS][BLOCKS + 1];
  __shared__ float lse[BLOCKS];

  const int b = blockIdx.x;
  const int t = threadIdx.x;           // 0..511
  const float invN = 1.0f / (float)BSIZE;

  // ---- Stage A: finish the mean pool into LDS, transposing [e,blk]->[blk,e]
  if (split > 0) {
#pragma unroll
    for (int i = 0; i < 8; ++i) {
      const int pos = t + i * 512;
      const int ei = pos >> 6, j = pos & 63;
      float aq = 0.f, ak = 0.f;
      for (int sp = 0; sp < split; ++sp) {
        aq += partial[((size_t)(b * 2 + 0) * split + sp) * POS + pos];
        ak += partial[((size_t)(b * 2 + 1) * split + sp) * POS + pos];
      }
      sq[j][ei] = aq * invN;
      sk[j][ei] = ak * invN;
    }
  } else {  // no-workspace fallback: pool directly from global
#pragma unroll
    for (int i = 0; i < 8; ++i) {
      const int pos = t + i * 512;
      const float* pq = b_q + (size_t)b * BSIZE * POS + pos;
      const float* pk = b_k + (size_t)b * BSIZE * POS + pos;
      float aq = 0.f, ak = 0.f;
      for (int s = 0; s < BSIZE; ++s) {
        aq += __builtin_nontemporal_load(pq + (size_t)s * POS);
        ak += __builtin_nontemporal_load(pk + (size_t)s * POS);
      }
      const int ei = pos >> 6, j = pos & 63;
      sq[j][ei] = aq * invN;
      sk[j][ei] = ak * invN;
    }
  }
  __syncthreads();

  // ---- Stage B: R = sq @ sk^T via V_WMMA_F32_16X16X4_F32 (native f32 WMMA)
  const int lane = t & 31;
  const int wave = t >> 5;                // 0..15
  const int tm   = (wave >> 2) * 16;      // C tile row base
  const int tn   = (wave & 3) * 16;       // C tile col base
  const int half = lane >> 4;             // 0: K pair {0,1}, 1: K pair {2,3}
  const int l    = lane & 15;             // M index (A) / N index (B)

  v8f c = {};
  for (int kk = 0; kk < ESIZE; kk += 4) {
    v2f a, bb;
    a.x  = sq[tm + l][kk + 2 * half];
    a.y  = sq[tm + l][kk + 2 * half + 1];
    bb.x = sk[tn + l][kk + 2 * half];      // B[k][n] = sk[n][k]
    bb.y = sk[tn + l][kk + 2 * half + 1];
    c = __builtin_amdgcn_wmma_f32_16x16x4_f32(
        /*neg_a=*/false, a, /*neg_b=*/false, bb,
        /*c_mod=*/(short)0, c, /*reuse_a=*/false, /*reuse_b=*/false);
  }

  // ---- Stage C: scale, relu, gumbel, log -> r in LDS
#pragma unroll
  for (int v = 0; v < 8; ++v) {
    const int i = tm + v + 8 * half;      // C layout: VGPR v -> row v (+8 hi)
    const int j = tn + l;
    float R = fmaxf(c[v] * SCALE, 0.0f);
    const float u = gumbel_u[((size_t)b * BLOCKS + i) * BLOCKS + j];
    const float g = -__logf(-__logf(u + EPS) + EPS);
    r[i][j] = (__logf(R) + g) * TEMP_INV; // log(0) -> -inf, exp() -> 0: ok
  }
  __syncthreads();

  // ---- Stage D: 8 Sinkhorn iterations (row lse then col lse), all in LDS
  for (int it = 0; it < 8; ++it) {
    if (t < BLOCKS) {                     // row logsumexp (over j)
      float m = -INFINITY;
      for (int j = 0; j < BLOCKS; ++j) m = fmaxf(m, r[t][j]);
      float s = 0.f;
      for (int j = 0; j < BLOCKS; ++j) s += __expf(r[t][j] - m);
      lse[t] = m + __logf(s);
    }
    __syncthreads();
#pragma unroll
    for (int i2 = 0; i2 < 8; ++i2) {
      const int idx = t + i2 * 512;
      r[idx >> 6][idx & 63] -= lse[idx >> 6];
    }
    __syncthreads();
    if (t < BLOCKS) {                     // col logsumexp (over i)
      float m = -INFINITY;
      for (int i = 0; i < BLOCKS; ++i) m = fmaxf(m, r[i][t]);
      float s = 0.f;
      for (int i = 0; i < BLOCKS; ++i) s += __expf(r[i][t] - m);
      lse[t] = m + __logf(s);
    }
    __syncthreads();
#pragma unroll
    for (int i2 = 0; i2 < 8; ++i2) {
      const int idx = t + i2 * 512;
      r[idx >> 6][idx & 63] -= lse[idx & 63];
    }
    __syncthreads();
  }

  // ---- Stage E: out = exp(r)
#pragma unroll
  for (int i2 = 0; i2 < 8; ++i2) {
    const int idx = t + i2 * 512;
    const int i = idx >> 6, j = idx & 63;
    out[((size_t)b * BLOCKS + i) * BLOCKS + j] = __expf(r[i][j]);
  }
}

// ---------------------------------------------------------------------------
extern "C" void kernel_launch(void* const* d_in, const int* in_sizes, int n_in,
                              void* d_out, int out_size, void* d_ws, size_t ws_size,
                              hipStream_t stream) {
  const float* b_q      = (const float*)d_in[0];
  const float* b_k      = (const float*)d_in[1];
  const float* gumbel_u = (const float*)d_in[2];
  float* out            = (float*)d_out;

  const int B = in_sizes[0] / (BSIZE * POS);   // 64

  // choose split factor by available scratch (deterministic: ws_size is fixed)
  const size_t per_split = (size_t)B * 2 * POS * sizeof(float);
  int split = 0;
  if      (ws_size >= 4 * per_split) split = 4;
  else if (ws_size >= 2 * per_split) split = 2;
  else if (ws_size >= 1 * per_split) split = 1;

  float* partial = (float*)d_ws;

  if (split > 0) {
    perm_pool_partial<<<dim3(B, 2, split), 256, 0, stream>>>(b_q, b_k, partial, split);
  }
  perm_sinkhorn<<<dim3(B), 512, 0, stream>>>(b_q, b_k, partial, split, gumbel_u, out);
}